// MultiscaleDeformableAttention_16372415333055
// MI455X (gfx1250) — compile-verified
//
#include <hip/hip_runtime.h>
#include <hip/hip_bf16.h>

typedef __attribute__((ext_vector_type(2))) float v2f;
typedef __attribute__((ext_vector_type(8))) float v8f;

static __device__ __forceinline__ v8f wmma4(v2f a, v2f b, v8f c) {
    // D = A(16x4,f32) x B(4x16,f32) + C(16x16,f32); full fp32 path.
    return __builtin_amdgcn_wmma_f32_16x16x4_f32(
        /*neg_a=*/false, a, /*neg_b=*/false, b,
        /*c_mod=*/(short)0, c, /*reuse_a=*/false, /*reuse_b=*/false);
}

// Generic fp32 WMMA GEMM: C[M,N] = A[M,Kd] * B[Kd,N] + bias[N]
// A row-major (lda), B row-major (ldb), C row-major (ldc).
// One wave computes a 64(M) x 32(N) tile of C: 8 WMMA accumulators,
// 8 WMMA per k-step against 8 vmem issues (4x b64 A + 4x b32 B).
// Requires: M%64==0, N%32==0, Kd%4==0.
__global__ __launch_bounds__(256) void gemm_f32_wmma_kernel(
    const float* __restrict__ A, int lda,
    const float* __restrict__ B, int ldb,
    const float* __restrict__ bias,
    float* __restrict__ C, int ldc,
    int M, int N, int Kd) {
    const int lane = threadIdx.x & 31;
    const int wave = threadIdx.x >> 5;
    const int ntn = N >> 5;
    const int ntiles = (M >> 6) * ntn;
    const int tile = blockIdx.x * 8 + wave;
    if (tile >= ntiles) return;            // wave-uniform: EXEC stays all-1s
    const int tm = tile / ntn;
    const int tn = tile - tm * ntn;

    const int l15 = lane & 15;             // M index (A) / N index (B,C)
    const int khalf = (lane >> 4) << 1;    // 0 or 2: which K-pair this half-wave holds

    const float* Arow[4];
    #pragma unroll
    for (int i = 0; i < 4; ++i)
        Arow[i] = A + (size_t)(tm * 64 + i * 16 + l15) * lda;
    const float* Bcol = B + (size_t)tn * 32 + l15;

    v8f c0[4] = {{}, {}, {}, {}};          // N-cols 0..15
    v8f c1[4] = {{}, {}, {}, {}};          // N-cols 16..31
    #pragma unroll 4
    for (int kk = 0; kk < Kd; kk += 4) {
        v2f a[4];
        #pragma unroll
        for (int i = 0; i < 4; ++i)
            a[i] = *(const v2f*)(Arow[i] + kk + khalf);
        const float* b0p = Bcol + (size_t)(kk + khalf) * ldb;
        const float* b1p = b0p + ldb;
        v2f b0 = { b0p[0],  b1p[0]  };
        v2f b1 = { b0p[16], b1p[16] };
        #pragma unroll
        for (int i = 0; i < 4; ++i) {
            c0[i] = wmma4(a[i], b0, c0[i]);
            c1[i] = wmma4(a[i], b1, c1[i]);
        }
    }

    const int halfm = (lane >> 4) << 3;    // 0 or 8
    const float bias0 = bias[tn * 32 + l15];
    const float bias1 = bias[tn * 32 + 16 + l15];
    float* Crow = C + (size_t)(tm * 64) * ldc + tn * 32;
    #pragma unroll
    for (int i = 0; i < 4; ++i) {
        #pragma unroll
        for (int r = 0; r < 8; ++r) {
            const int m0 = i * 16 + r + halfm;   // row within 64-row tile
            float* p = Crow + (size_t)m0 * ldc;
            p[l15]      = c0[i][r] + bias0;
            p[16 + l15] = c1[i][r] + bias1;
        }
    }
}

static __device__ __forceinline__ float dev_sigmoid(float x) {
    return 1.0f / (1.0f + expf(-x));
}

// Per (b,q): softmax attn logits (in place, per head over 16 values) and
// convert raw offsets into sampling locations (in place).
__global__ __launch_bounds__(256) void prep_kernel(
    const float* __restrict__ geom,
    float* __restrict__ attn,   // [BQ,128]  logits -> softmax
    float* __restrict__ off,    // [BQ,256]  raw offsets -> loc (x,y pairs)
    int BQ) {
    const int bq = blockIdx.x * blockDim.x + threadIdx.x;
    if (bq >= BQ) return;

    float* arow = attn + (size_t)bq * 128;
    #pragma unroll
    for (int h = 0; h < 8; ++h) {
        float vals[16];
        float mx = -3.0e38f;
        #pragma unroll
        for (int i = 0; i < 16; ++i) { vals[i] = arow[h * 16 + i]; mx = fmaxf(mx, vals[i]); }
        float s = 0.0f;
        #pragma unroll
        for (int i = 0; i < 16; ++i) { vals[i] = expf(vals[i] - mx); s += vals[i]; }
        const float inv = 1.0f / s;
        #pragma unroll
        for (int i = 0; i < 16; ++i) arow[h * 16 + i] = vals[i] * inv;
    }

    const float* g = geom + (size_t)bq * 4;
    const float cx = dev_sigmoid(g[0]);
    const float cy = dev_sigmoid(g[1]);
    const float sx = dev_sigmoid(g[2]) * 0.125f;  // wh * (1/K=0.25) * OFFSET_SCALE=0.5
    const float sy = dev_sigmoid(g[3]) * 0.125f;

    float* orow = off + (size_t)bq * 256;
    for (int i = 0; i < 128; ++i) {
        const float ox = orow[2 * i];
        const float oy = orow[2 * i + 1];
        orow[2 * i]     = cx + ox * sx;
        orow[2 * i + 1] = cy + oy * sy;
    }
}

static __device__ __forceinline__ float fetchv(const float* __restrict__ vb,
                                               int xi, int yi, int Wl) {
    if (xi < 0 || xi >= Wl || yi < 0 || yi >= Wl) return 0.0f;
    return vb[(size_t)(yi * Wl + xi) * 256];
}

// One block per (b,q,h); thread d owns channel d of the 256-dim output.
__global__ __launch_bounds__(256) void sample_kernel(
    const float* __restrict__ v_all,
    const float* __restrict__ attn,   // [BQ,128] softmaxed
    const float* __restrict__ loc,    // [BQ,256] (h,l,k,2)
    float* __restrict__ out_mid,      // [BQ, H*D] = [BQ,2048]
    int Q, int B) {
    const int idx = blockIdx.x;       // bq*8 + h
    const int d = threadIdx.x;
    const int h = idx & 7;
    const int bq = idx >> 3;
    const int b = bq / Q;

    const int widths[4] = {64, 32, 16, 8};
    const int cumHW[4]  = {0, 4096, 5120, 5376};

    const float* arow = attn + (size_t)bq * 128 + h * 16;
    const float* lrow = loc  + (size_t)bq * 256 + h * 32;

    float acc = 0.0f;
    #pragma unroll
    for (int l = 0; l < 4; ++l) {
        const int Wl = widths[l];
        const int HW = Wl * Wl;
        const float* vb = v_all + ((size_t)B * cumHW[l] + (size_t)b * HW) * 256 + d;
        #pragma unroll
        for (int k = 0; k < 4; ++k) {
            const float aw = arow[l * 4 + k];
            const float lx = lrow[l * 8 + k * 2];
            const float ly = lrow[l * 8 + k * 2 + 1];
            // grid_sample, align_corners=False, zeros padding
            const float gx = fminf(fmaxf(2.0f * lx - 1.0f, -1.0f), 1.0f);
            const float gy = fminf(fmaxf(2.0f * ly - 1.0f, -1.0f), 1.0f);
            const float x = (gx + 1.0f) * 0.5f * (float)Wl - 0.5f;
            const float y = (gy + 1.0f) * 0.5f * (float)Wl - 0.5f;
            const float x0f = floorf(x);
            const float y0f = floorf(y);
            const float wx = x - x0f;
            const float wy = y - y0f;
            const int x0 = (int)x0f, y0 = (int)y0f;
            const float w00 = (1.0f - wx) * (1.0f - wy);
            const float w01 = wx * (1.0f - wy);
            const float w10 = (1.0f - wx) * wy;
            const float w11 = wx * wy;
            float s = w00 * fetchv(vb, x0,     y0,     Wl)
                    + w01 * fetchv(vb, x0 + 1, y0,     Wl)
                    + w10 * fetchv(vb, x0,     y0 + 1, Wl)
                    + w11 * fetchv(vb, x0 + 1, y0 + 1, Wl);
            acc += aw * s;
        }
    }
    out_mid[(size_t)idx * 256 + d] = acc;
}

// bsum[e] = sum_h b_final[h][e]
__global__ __launch_bounds__(256) void bsum_kernel(const float* __restrict__ b_final,
                                                   float* __restrict__ bsum) {
    const int e = threadIdx.x;
    float s = 0.0f;
    #pragma unroll
    for (int h = 0; h < 8; ++h) s += b_final[h * 256 + e];
    bsum[e] = s;
}

extern "C" void kernel_launch(void* const* d_in, const int* in_sizes, int n_in,
                              void* d_out, int out_size, void* d_ws, size_t ws_size,
                              hipStream_t stream) {
    const float* queries = (const float*)d_in[0];   // [B,Q,256]
    const float* geom    = (const float*)d_in[1];   // [B,Q,4]
    const float* value_l[4] = { (const float*)d_in[2], (const float*)d_in[3],
                                (const float*)d_in[4], (const float*)d_in[5] };
    const float* W_off   = (const float*)d_in[6];   // [256,256]
    const float* b_off   = (const float*)d_in[7];   // [256]
    const float* W_attn  = (const float*)d_in[8];   // [256,128]
    const float* b_attn  = (const float*)d_in[9];   // [128]
    const float* W_val   = (const float*)d_in[10];  // [256,256]
    const float* b_val   = (const float*)d_in[11];  // [256]
    const float* W_final = (const float*)d_in[12];  // [8,256,256] == [2048,256] row-major
    const float* b_final = (const float*)d_in[13];  // [8,256]
    float* y = (float*)d_out;                       // [B,Q,256]

    const int D = 256;
    const int BQ = in_sizes[0] / D;   // B*Q = 4096
    const int Q = 2048;
    const int B = BQ / Q;             // 2
    const int HW[4] = {4096, 1024, 256, 64};
    const int cumHW[4] = {0, 4096, 5120, 5376};

    // Workspace layout (floats), all 1KB-aligned regions
    float* ws = (float*)d_ws;
    float* v_all   = ws;                          // B*5440*256 = 2,785,280
    float* attn_ws = ws + 2785280;                // BQ*128     =   524,288
    float* off_ws  = ws + 3309568;                // BQ*256     = 1,048,576
    float* out_mid = ws + 4358144;                // BQ*2048    = 8,388,608
    float* bsum    = ws + 12746752;               // 256

    dim3 blk(256);
    auto grid_for = [](int M, int N) { int t = (M / 64) * (N / 32); return dim3((t + 7) / 8); };

    // 1) attn logits = queries @ W_attn + b_attn   [BQ x 128]
    gemm_f32_wmma_kernel<<<grid_for(BQ, 128), blk, 0, stream>>>(
        queries, 256, W_attn, 128, b_attn, attn_ws, 128, BQ, 128, 256);

    // 2) off = queries @ W_off + b_off             [BQ x 256]
    gemm_f32_wmma_kernel<<<grid_for(BQ, 256), blk, 0, stream>>>(
        queries, 256, W_off, 256, b_off, off_ws, 256, BQ, 256, 256);

    // 3) softmax(attn) in place; loc from geometry + off in place
    prep_kernel<<<dim3((BQ + 255) / 256), blk, 0, stream>>>(geom, attn_ws, off_ws, BQ);

    // 4) v_l = value_l @ W_val + b_val per level into v_all
    for (int l = 0; l < 4; ++l) {
        const int M = B * HW[l];
        gemm_f32_wmma_kernel<<<grid_for(M, 256), blk, 0, stream>>>(
            value_l[l], 256, W_val, 256, b_val,
            v_all + (size_t)B * cumHW[l] * 256, 256, M, 256, 256);
    }

    // 5) bias reduction for the final projection
    bsum_kernel<<<dim3(1), blk, 0, stream>>>(b_final, bsum);

    // 6) bilinear sampling + attention accumulation -> out_mid [BQ, H*D]
    sample_kernel<<<dim3(BQ * 8), blk, 0, stream>>>(v_all, attn_ws, off_ws, out_mid, Q, B);

    // 7) y = out_mid [BQ x 2048] @ W_final [2048 x 256] + bsum  (the 4.3 GFLOP GEMM)
    gemm_f32_wmma_kernel<<<grid_for(BQ, 256), blk, 0, stream>>>(
        out_mid, 2048, W_final, 256, bsum, y, 256, BQ, 256, 2048);
}